// TransformerLoss_34926674051377
// MI455X (gfx1250) — compile-verified
//
#include <hip/hip_runtime.h>
#include <hip/hip_bf16.h>

// ---------------------------------------------------------------------------
// TransformerLoss (Tacotron2-style): masked mel L1 + gate BCE + guided-attn.
// Memory-bound: ~90 MB of reads -> ~3.9 us at 23.3 TB/s. Strategy:
//   * single-pass coalesced float4 streaming; read ONLY the 4 needed
//     (layer,head) alignment planes out of 24,
//   * share the exp() guide weight across the 4 planes (4x fewer trans ops),
//   * 32-bit indexing everywhere (all offsets < 2^31) so addressing lowers
//     to SGPR-base + voffset instead of 64-bit VALU adds,
//   * hierarchical reduction: thread -> wave (V_WMMA_F32_16X16X4_F32 with
//     ones-B = exact f32 row sums, + one ds_swizzle SWAPX16) -> block LDS
//     (8 waves, fixed) -> one atomicAdd per block.
// ---------------------------------------------------------------------------

typedef __attribute__((ext_vector_type(2))) float v2f;
typedef __attribute__((ext_vector_type(8))) float v8f;

#define B_SZ     32
#define N_MEL_   80
#define T_MEL_   800
#define T_TEXT_  160
#define POS_W_   5.0f
#define GUIDE_K_ 3.125f
#define BLOCK_   256   // 8 waves on wave32

// Full-wave (32 lane) sum on the f32 WMMA unit.
// A (16x4): lane L<16 -> row M=L {K0,K1}; lane>=16 -> row M=L-16 {K2,K3}.
// A vgpr0 = v, vgpr1 = 0; B = all ones (layout-invariant).
//   => D[m][n] = v(m) + v(m+16) for all n.
// D layout: lane L<16 holds M=0..7 across its 8 VGPRs, lane>=16 holds M=8..15,
// so (sum of own 8 VGPRs) + (same from lane^16 via ds_swizzle SWAPX16) = total.
__device__ __forceinline__ float wave_reduce_sum_wmma(float v) {
  v2f a; a[0] = v;    a[1] = 0.0f;
  v2f b; b[0] = 1.0f; b[1] = 1.0f;
  v8f c = {};
  c = __builtin_amdgcn_wmma_f32_16x16x4_f32(
      /*neg_a=*/false, a, /*neg_b=*/false, b,
      /*c_mod=*/(short)0, c, /*reuse_a=*/false, /*reuse_b=*/false);
  float s = ((c[0] + c[1]) + (c[2] + c[3])) + ((c[4] + c[5]) + (c[6] + c[7]));
  // SWAPX16: offset[14:10]=xor=0x10, or=0, and_mask=0x1f -> 0x401F
  s += __int_as_float(__builtin_amdgcn_ds_swizzle(__float_as_int(s), 0x401F));
  return s;
}

// Block (256-thread) sum of v, then one atomicAdd into *dst.
// Call with all threads active (WMMA needs EXEC all-1s).
__device__ __forceinline__ void block_reduce_atomic_256(float v, float* dst) {
  __shared__ float wsum[8];
  float s = wave_reduce_sum_wmma(v);
  if ((threadIdx.x & 31u) == 0u) wsum[threadIdx.x >> 5] = s;
  __syncthreads();
  if (threadIdx.x == 0) {
    float t = ((wsum[0] + wsum[1]) + (wsum[2] + wsum[3])) +
              ((wsum[4] + wsum[5]) + (wsum[6] + wsum[7]));
    atomicAdd(dst, t);
  }
}

// --------------------------- mel L1 (two streams) ---------------------------
// sum over valid (b,c,t) of |mel_out - tgt| + |mel_post - tgt|
__global__ void __launch_bounds__(BLOCK_)
mel_l1_kernel(const float* __restrict__ mel_out,
              const float* __restrict__ mel_post,
              const float* __restrict__ mel_tgt,
              const int*   __restrict__ mel_len,
              float* __restrict__ acc) {
  const unsigned N4 = (unsigned)B_SZ * N_MEL_ * T_MEL_ / 4u;  // 512000 float4s
  const unsigned stride = gridDim.x * BLOCK_;
  float a = 0.0f;
  for (unsigned i = blockIdx.x * BLOCK_ + threadIdx.x; i < N4; i += stride) {
    unsigned e  = i << 2;                     // element index, 4-aligned
    int t0 = (int)(e % (unsigned)T_MEL_);     // T_MEL_ % 4 == 0 -> same row
    int b  = (int)(e / (unsigned)(N_MEL_ * T_MEL_));
    int ml = mel_len[b];
    float4 mo = ((const float4*)mel_out)[i];
    float4 mp = ((const float4*)mel_post)[i];
    float4 mt = ((const float4*)mel_tgt)[i];
    float v0 = (t0 + 0 < ml) ? 1.0f : 0.0f;
    float v1 = (t0 + 1 < ml) ? 1.0f : 0.0f;
    float v2 = (t0 + 2 < ml) ? 1.0f : 0.0f;
    float v3 = (t0 + 3 < ml) ? 1.0f : 0.0f;
    a += v0 * (fabsf(mo.x - mt.x) + fabsf(mp.x - mt.x));
    a += v1 * (fabsf(mo.y - mt.y) + fabsf(mp.y - mt.y));
    a += v2 * (fabsf(mo.z - mt.z) + fabsf(mp.z - mt.z));
    a += v3 * (fabsf(mo.w - mt.w) + fabsf(mp.w - mt.w));
  }
  block_reduce_atomic_256(a, acc);
}

// ------------------------------ gate BCE ------------------------------------
// softplus(x) = max(x,0) + log1p(exp(-|x|))  (stable)
__global__ void __launch_bounds__(BLOCK_)
gate_bce_kernel(const float* __restrict__ x,
                const float* __restrict__ y,
                const int*   __restrict__ mel_len,
                float* __restrict__ acc) {
  const unsigned N = (unsigned)B_SZ * T_MEL_;          // 25600
  const unsigned stride = gridDim.x * BLOCK_;
  float a = 0.0f;
  for (unsigned i = blockIdx.x * BLOCK_ + threadIdx.x; i < N; i += stride) {
    int b = (int)(i / (unsigned)T_MEL_);
    int t = (int)(i % (unsigned)T_MEL_);
    if (t < mel_len[b]) {
      float xv = x[i], yv = y[i];
      float l1pe  = log1pf(expf(-fabsf(xv)));
      float sp_neg = fmaxf(-xv, 0.0f) + l1pe;   // softplus(-x)
      float sp_pos = fmaxf( xv, 0.0f) + l1pe;   // softplus(x)
      a += POS_W_ * yv * sp_neg + (1.0f - yv) * sp_pos;
    }
  }
  block_reduce_atomic_256(a, acc);
}

// --------------------------- guided attention -------------------------------
// sum over b, layers {4,5}, heads {0,1}, (t,j) of  A[...] * W(b,t,j).
// One thread per (b,t,j0..j0+3); W computed once, shared by all 4 planes.
// Plane base pointers (p0..p3) hoisted to arguments; in-plane offset is a
// 32-bit voffset (max 95.3M floats < 2^31).
__global__ void __launch_bounds__(BLOCK_)
guide_kernel(const float* __restrict__ p0,   // align + (4*nH+0)*Tm*Lt
             const float* __restrict__ p1,   // align + (4*nH+1)*Tm*Lt
             const float* __restrict__ p2,   // align + (5*nH+0)*Tm*Lt
             const float* __restrict__ p3,   // align + (5*nH+1)*Tm*Lt
             const int* __restrict__ text_len,
             const int* __restrict__ mel_len,
             float* __restrict__ acc) {
  const unsigned rowQ = T_TEXT_ / 4u;                     // 40 float4s / (b,t)
  const unsigned N    = (unsigned)B_SZ * T_MEL_ * rowQ;   // 1,024,000
  const unsigned SB   = 6u * 4u * T_MEL_ * T_TEXT_;       // 3,072,000 (batch)
  const unsigned stride = gridDim.x * BLOCK_;
  float a = 0.0f;
  for (unsigned i = blockIdx.x * BLOCK_ + threadIdx.x; i < N; i += stride) {
    unsigned q  = i % rowQ;
    unsigned t  = (i / rowQ) % (unsigned)T_MEL_;
    unsigned b  = i / (rowQ * (unsigned)T_MEL_);
    int j0 = (int)(q * 4u);
    int ml = mel_len[b];
    int tl = text_len[b];
    unsigned idx = b * SB + t * (unsigned)T_TEXT_ + (unsigned)j0;
    float4 q0 = *(const float4*)(p0 + idx);
    float4 q1 = *(const float4*)(p1 + idx);
    float4 q2 = *(const float4*)(p2 + idx);
    float4 q3 = *(const float4*)(p3 + idx);
    float sj[4] = { q0.x + q1.x + q2.x + q3.x,
                    q0.y + q1.y + q2.y + q3.y,
                    q0.z + q1.z + q2.z + q3.z,
                    q0.w + q1.w + q2.w + q3.w };
    float inv_ml = 1.0f / (float)ml;
    float inv_tl = 1.0f / (float)tl;
    float tn = (float)(int)t * inv_ml;
    float tv = ((int)t < ml) ? 1.0f : 0.0f;
#pragma unroll
    for (int k = 0; k < 4; ++k) {
      int j = j0 + k;
      float d = tn - (float)j * inv_tl;
      float W = 1.0f - __expf(-GUIDE_K_ * d * d);
      W = (j < tl) ? W : 0.0f;
      a += sj[k] * (W * tv);
    }
  }
  block_reduce_atomic_256(a, acc);
}

// ------------------------------- finalize -----------------------------------
// Mask is rectangular: n = sum_b ml[b]; gmask_sum = sum_b ml[b]*tl[b].
__global__ void finalize_kernel(const int* __restrict__ text_len,
                                const int* __restrict__ mel_len,
                                const float* __restrict__ ws,
                                float* __restrict__ out) {
  if (threadIdx.x == 0 && blockIdx.x == 0) {
    int n = 0; long long g = 0;
    for (int b = 0; b < B_SZ; ++b) {
      int ml = min(mel_len[b], T_MEL_);
      int tl = min(text_len[b], T_TEXT_);
      n += ml;
      g += (long long)ml * tl;
    }
    out[0] = ws[0] / ((float)n * (float)N_MEL_);  // mel = (l1a+l1b)/n_mel
    out[1] = ws[1] / (float)n;                    // bce
    out[2] = ws[2] / (4.0f * (float)g);           // guide
  }
}

extern "C" void kernel_launch(void* const* d_in, const int* in_sizes, int n_in,
                              void* d_out, int out_size, void* d_ws, size_t ws_size,
                              hipStream_t stream) {
  const float* mel_out  = (const float*)d_in[0];
  const float* mel_post = (const float*)d_in[1];
  const float* gate_out = (const float*)d_in[2];
  const float* mel_tgt  = (const float*)d_in[3];
  const float* gate_tgt = (const float*)d_in[4];
  const float* align    = (const float*)d_in[5];
  const int*   text_len = (const int*)d_in[6];
  const int*   mel_len  = (const int*)d_in[7];
  float* acc = (float*)d_ws;   // acc[0]=mel L1 sum, acc[1]=bce sum, acc[2]=guide sum
  float* out = (float*)d_out;

  hipMemsetAsync(acc, 0, 3 * sizeof(float), stream);

  // alignments (B, 6, 4, 800, 160): planes for layers {4,5} x heads {0,1}
  const unsigned SH = (unsigned)T_MEL_ * T_TEXT_;   // 128,000 head stride
  const float* p0 = align + (4u * 4u + 0u) * SH;    // layer4 head0
  const float* p1 = align + (4u * 4u + 1u) * SH;    // layer4 head1
  const float* p2 = align + (5u * 4u + 0u) * SH;    // layer5 head0
  const float* p3 = align + (5u * 4u + 1u) * SH;    // layer5 head1

  mel_l1_kernel  <<<2048, BLOCK_, 0, stream>>>(mel_out, mel_post, mel_tgt, mel_len, acc + 0);
  gate_bce_kernel<<< 100, BLOCK_, 0, stream>>>(gate_out, gate_tgt, mel_len, acc + 1);
  guide_kernel   <<<4096, BLOCK_, 0, stream>>>(p0, p1, p2, p3, text_len, mel_len, acc + 2);
  finalize_kernel<<<   1,  32,    0, stream>>>(text_len, mel_len, acc, out);
}